// RegionIntegrator_39204461478392
// MI455X (gfx1250) — compile-verified
//
#include <hip/hip_runtime.h>
#include <stdint.h>

// RegionIntegrator: overlap-add of 225 (64x64) regions on a 32-stride grid into
// (4,16,512,512), normalized by cover count. Pure bandwidth problem (~300 MB).
// CDNA5 path: async global->LDS DMA (ASYNCcnt) with double buffering.

namespace {
constexpr int kB          = 4;
constexpr int kC          = 16;
constexpr int kH          = 512;
constexpr int kRS         = 64;
constexpr int kStep       = 32;
constexpr int kGrid       = 15;               // regions per dim
constexpr int kNReg       = kGrid * kGrid;    // 225
constexpr int kTile       = 32;               // output tile side
constexpr int kTilesPD    = kH / kTile;       // 16
constexpr int kTileFloats = kTile * kTile;    // 1024 floats = 4 KB
constexpr int kSlotBytes  = kTileFloats * 4;  // 4096
constexpr int kBufBytes   = 4 * kSlotBytes;   // 16384 per channel-slice buffer
}

__global__ __launch_bounds__(256) void region_integrate_kernel(
    const float* __restrict__ regions, float* __restrict__ out)
{
  const int t  = threadIdx.x;   // 0..255
  const int r  = t >> 3;        // tile row 0..31
  const int s8 = t & 7;         // 16-byte chunk within row, 0..7

  const int blk = blockIdx.x;   // b*256 + ty*16 + tx
  const int tx  = blk & 15;
  const int ty  = (blk >> 4) & 15;
  const int b   = blk >> 8;

  const int y0 = ty * kTile;
  const int x0 = tx * kTile;

  // Clamped covering region indices. Duplicates at edges are intentional:
  // duplication factor (2/cnt_y)*(2/cnt_x) makes sum*0.25 == sum/(cnt_y*cnt_x).
  const int is[2] = { (ty > 0) ? ty - 1 : 0, (ty < kGrid) ? ty : kGrid - 1 };
  const int js[2] = { (tx > 0) ? tx - 1 : 0, (tx < kGrid) ? tx : kGrid - 1 };

  __shared__ float smem[2 * 4 * kTileFloats];   // 32 KB, dual-buffered
  // Low 32 bits of a flat LDS pointer are the LDS byte offset.
  const unsigned lds0 = (unsigned)(uintptr_t)smem + (unsigned)(t * 16);

  // Per-slot global byte addresses for channel 0; channel stride = 16 KB.
  uint64_t ga[4];
#pragma unroll
  for (int si = 0; si < 2; ++si) {
#pragma unroll
    for (int sj = 0; sj < 2; ++sj) {
      const int n  = is[si] * kGrid + js[sj];
      const int ry = y0 - kStep * is[si] + r;        // 0..63
      const int rx = x0 - kStep * js[sj] + s8 * 4;   // 0..60, 16B aligned
      const long long foff =
          ((long long)(b * kNReg + n) * kC) * (long long)(kRS * kRS)
          + (long long)(ry * kRS + rx);
      ga[si * 2 + sj] = (uint64_t)(uintptr_t)(regions + foff);
    }
  }

  auto issue = [&](int buf) {
#pragma unroll
    for (int s = 0; s < 4; ++s) {
      const unsigned ldsa =
          lds0 + (unsigned)(buf * kBufBytes + s * kSlotBytes);
      // GV mode: vdst = LDS byte offset, vaddr = 64-bit global address.
      asm volatile("global_load_async_to_lds_b128 %0, %1, off"
                   :: "v"(ldsa), "v"(ga[s]) : "memory");
    }
  };

  issue(0);  // prologue: channel 0 into buffer 0

  for (int c = 0; c < kC; ++c) {
    if (c + 1 < kC) {
#pragma unroll
      for (int s = 0; s < 4; ++s) ga[s] += (uint64_t)(kRS * kRS * 4);
      // WAR guard: make sure prior ds reads of the buffer we are about to
      // overwrite have drained before the DMA can touch it.
      asm volatile("s_wait_dscnt 0x0" ::: "memory");
      issue((c + 1) & 1);
      // Retire the 4 loads of the current buffer; keep 4 in flight.
      asm volatile("s_wait_asynccnt 0x4" ::: "memory");
    } else {
      asm volatile("s_wait_asynccnt 0x0" ::: "memory");
    }

    // Each lane consumes exactly the chunk its DMA wrote (no cross-lane
    // sharing -> no workgroup barrier needed, ASYNCcnt wait suffices).
    const float4* lp =
        reinterpret_cast<const float4*>(smem + (c & 1) * 4 * kTileFloats) + t;
    const float4 v0 = lp[0];
    const float4 v1 = lp[256];
    const float4 v2 = lp[512];
    const float4 v3 = lp[768];

    float4 acc;
    acc.x = 0.25f * ((v0.x + v1.x) + (v2.x + v3.x));
    acc.y = 0.25f * ((v0.y + v1.y) + (v2.y + v3.y));
    acc.z = 0.25f * ((v0.z + v1.z) + (v2.z + v3.z));
    acc.w = 0.25f * ((v0.w + v1.w) + (v2.w + v3.w));

    const long long ooff =
        (((long long)(b * kC + c) * kH) + (long long)(y0 + r)) * kH
        + (long long)(x0 + s8 * 4);
    *reinterpret_cast<float4*>(out + ooff) = acc;
  }
}

extern "C" void kernel_launch(void* const* d_in, const int* in_sizes, int n_in,
                              void* d_out, int out_size, void* d_ws, size_t ws_size,
                              hipStream_t stream) {
  (void)in_sizes; (void)n_in; (void)d_ws; (void)ws_size; (void)out_size;
  const float* regions = (const float*)d_in[0];  // (4,225,16,64,64) f32
  float* out = (float*)d_out;                    // (4,16,512,512) f32
  dim3 grid(kB * kTilesPD * kTilesPD);           // 1024 blocks
  dim3 block(256);                               // 8 wave32
  hipLaunchKernelGGL(region_integrate_kernel, grid, block, 0, stream,
                     regions, out);
}